// ResidualVectorQuantize_1726576854454
// MI455X (gfx1250) — compile-verified
//
#include <hip/hip_runtime.h>
#include <math.h>
#include <stdint.h>

// Problem constants (from reference setup_inputs)
#define RVQ_B  8
#define RVQ_D  512
#define RVQ_T  4096
#define RVQ_N  9
#define RVQ_CS 1024
#define RVQ_CD 8

typedef float v2f __attribute__((ext_vector_type(2)));
typedef float v8f __attribute__((ext_vector_type(8)));

__device__ __forceinline__ v8f wmma4(v2f a, v2f b, v8f c) {
    // V_WMMA_F32_16X16X4_F32 : D = A(16x4) * B(4x16) + C(16x16), all fp32
    return __builtin_amdgcn_wmma_f32_16x16x4_f32(false, a, false, b, (short)0, c,
                                                 false, false);
}

// CDNA5 async global->LDS DMA (ASYNCcnt-tracked); inline asm is portable across
// both toolchains (bypasses the builtin arity mismatch).
__device__ __forceinline__ void async_lds_b128(uint32_t lds_off, const void* g) {
    asm volatile("global_load_async_to_lds_b128 %0, %1, off"
                 :: "v"(lds_off), "v"((unsigned long long)(uintptr_t)g)
                 : "memory");
}
__device__ __forceinline__ void wait_async0() {
    asm volatile("s_wait_asynccnt 0x0" ::: "memory");
}

// ---------------- prep kernels ----------------

__global__ void rvq_zero_losses(float* loss) {
    if (threadIdx.x == 0) { loss[0] = 0.f; loss[1] = 0.f; }
}

// weight-norm rows of length 512, padded to 16 rows/stage (rows 8..15 zero)
// so the WMMA A-operand load needs no lane predication.
__global__ void rvq_prep_in(const float* __restrict__ v, const float* __restrict__ g,
                            float* __restrict__ w) {
    int row = blockIdx.x * blockDim.x + threadIdx.x;      // 0 .. N*16-1
    if (row >= RVQ_N * 16) return;
    int st = row >> 4, m = row & 15;
    float* wr = w + (size_t)row * RVQ_D;
    if (m >= RVQ_CD) {
        for (int d = 0; d < RVQ_D; ++d) wr[d] = 0.f;
        return;
    }
    const float* vr = v + ((size_t)st * RVQ_CD + m) * RVQ_D;
    float s = 0.f;
    for (int d = 0; d < RVQ_D; ++d) { float x = vr[d]; s += x * x; }
    float inv = g[st * RVQ_CD + m] / fmaxf(sqrtf(s), 1e-12f);
    for (int d = 0; d < RVQ_D; ++d) wr[d] = vr[d] * inv;
}

// weight-norm over rows of length 8 (w_out)
__global__ void rvq_prep_row8(const float* __restrict__ v, const float* __restrict__ g,
                              float* __restrict__ w, int rows) {
    int row = blockIdx.x * blockDim.x + threadIdx.x;
    if (row >= rows) return;
    const float* vr = v + (size_t)row * RVQ_CD;
    float s = 0.f;
    for (int k = 0; k < RVQ_CD; ++k) { float x = vr[k]; s += x * x; }
    float inv = g[row] / fmaxf(sqrtf(s), 1e-12f);
    float* wr = w + (size_t)row * RVQ_CD;
    for (int k = 0; k < RVQ_CD; ++k) wr[k] = vr[k] * inv;
}

// l2-normalize codebook rows, permuted into WMMA B-operand order
// {k0,k2,k1,k3,k4,k6,k5,k7} so each B operand is one 8B LDS load.
__global__ void rvq_prep_cbn(const float* __restrict__ v, float* __restrict__ w, int rows) {
    int row = blockIdx.x * blockDim.x + threadIdx.x;
    if (row >= rows) return;
    const float* vr = v + (size_t)row * RVQ_CD;
    float s = 0.f;
    for (int k = 0; k < RVQ_CD; ++k) { float x = vr[k]; s += x * x; }
    float inv = 1.f / fmaxf(sqrtf(s), 1e-12f);
    float* wr = w + (size_t)row * RVQ_CD;
    const int perm[8] = {0, 2, 1, 3, 4, 6, 5, 7};
    #pragma unroll
    for (int k = 0; k < RVQ_CD; ++k) wr[k] = vr[perm[k]] * inv;
}

// ---------------- fused per-stage kernel ----------------
// Block: 256 threads = 8 waves; wave owns 16 consecutive tokens (t).
// residual = z - z_q  (z_q accumulated in place; FIRST writes without read).

template <int FIRST>
__global__ __launch_bounds__(256) void rvq_stage(
    const float* __restrict__ z, float* __restrict__ zq,
    int* __restrict__ codes, float* __restrict__ lat, float* __restrict__ loss,
    const float* __restrict__ wn_in, const float* __restrict__ wn_out,
    const float* __restrict__ cbn, const float* __restrict__ cbraw,
    const float* __restrict__ in_b, const float* __restrict__ out_b,
    int stage)
{
    __shared__ __align__(16) float s_cbn[RVQ_CS * RVQ_CD];   // permuted cb_n (32 KB)
    __shared__ __align__(16) float s_wout[RVQ_D * RVQ_CD];   // w_out (16 KB)
    __shared__ float s_enc[8][RVQ_CD * 16];                  // per-wave enc_n
    __shared__ int   s_idx[8][16];                           // per-wave code indices

    const int tid  = threadIdx.x;
    const int wave = tid >> 5;
    const int lane = tid & 31;
    const int n16  = lane & 15;   // token/col within tile, or A-matrix row M
    const int half = lane >> 4;   // which K pair this lane supplies

    // ---- kick off async LDS staging; it overlaps the whole in_proj GEMM ----
    {
        const float* cbn_i = cbn + (size_t)stage * RVQ_CS * RVQ_CD;
        const float* wo_i  = wn_out + (size_t)stage * RVQ_D * RVQ_CD;
        uint32_t cb_lds = (uint32_t)(uintptr_t)&s_cbn[0];
        uint32_t wo_lds = (uint32_t)(uintptr_t)&s_wout[0];
        #pragma unroll
        for (int j = 0; j < 8; ++j) {                 // 2048 x 16B
            int e = tid + j * 256;
            async_lds_b128(cb_lds + e * 16, (const char*)cbn_i + (size_t)e * 16);
        }
        #pragma unroll
        for (int j = 0; j < 4; ++j) {                 // 1024 x 16B
            int e = tid + j * 256;
            async_lds_b128(wo_lds + e * 16, (const char*)wo_i + (size_t)e * 16);
        }
    }

    const int tile = blockIdx.x * 8 + wave;       // 0..2047
    const int bidx = tile >> 8;                   // batch
    const int t0   = (tile & 255) << 4;           // first token t of tile

    const float* zb = z  + (size_t)bidx * RVQ_D * RVQ_T;
    float*       qb = zq + (size_t)bidx * RVQ_D * RVQ_T;
    const float* Wi = wn_in + (size_t)stage * 16 * RVQ_D;   // padded 16x512

    // ---- in_proj: z_e = Wi(16x512) x residual(512x16), K in chunks of 4 ----
    v8f acc = {0.f,0.f,0.f,0.f,0.f,0.f,0.f,0.f};
    #pragma unroll 4
    for (int k0 = 0; k0 < RVQ_D; k0 += 4) {
        v2f a = *(const v2f*)&Wi[n16 * RVQ_D + k0 + 2 * half];  // (k, k+1) pair
        int kb = k0 + half;
        size_t o0 = (size_t)kb * RVQ_T + t0 + n16;
        size_t o1 = (size_t)(kb + 2) * RVQ_T + t0 + n16;
        float r0 = zb[o0], r1 = zb[o1];
        if (!FIRST) { r0 -= qb[o0]; r1 -= qb[o1]; }
        v2f bb; bb.x = r0; bb.y = r1;
        acc = wmma4(a, bb, acc);
    }

    // ---- bias, latents out, l2-normalize encodings ----
    float ze[RVQ_CD];
    if (lane < 16) {                 // VGPR r, lanes 0-15: cd=r, token=lane
        float ss = 0.f;
        #pragma unroll
        for (int r = 0; r < RVQ_CD; ++r) {
            float v = acc[r] + in_b[stage * RVQ_CD + r];
            ze[r] = v; ss += v * v;
            lat[((size_t)bidx * (RVQ_N * RVQ_CD) + stage * RVQ_CD + r) * RVQ_T + t0 + lane] = v;
        }
        float scl = 1.f / fmaxf(sqrtf(ss), 1e-12f);
        #pragma unroll
        for (int r = 0; r < RVQ_CD; ++r) s_enc[wave][r * 16 + lane] = ze[r] * scl;
    }
    wait_async0();        // drain this wave's async LDS writes
    __syncthreads();      // all waves' staging + enc stores visible

    // ---- distance GEMM: S = enc_n(16x8) x cbn^T(8x1024); argmax == nearest ----
    v2f A0, A1;
    A0.x = s_enc[wave][(2 * half + 0) * 16 + n16];
    A0.y = s_enc[wave][(2 * half + 1) * 16 + n16];
    A1.x = s_enc[wave][(2 * half + 4) * 16 + n16];
    A1.y = s_enc[wave][(2 * half + 5) * 16 + n16];

    float bestv[8]; int besti[8];
    #pragma unroll
    for (int r = 0; r < 8; ++r) { bestv[r] = -3.4e38f; besti[r] = 0; }

    #pragma unroll 2
    for (int c0 = 0; c0 < RVQ_CS; c0 += 16) {
        int code = c0 + n16;
        const float* bp = &s_cbn[code * RVQ_CD + 2 * half];   // permuted layout
        v2f B0 = *(const v2f*)bp;                             // one ds_load_b64
        v2f B1 = *(const v2f*)(bp + 4);
        v8f s = {0.f,0.f,0.f,0.f,0.f,0.f,0.f,0.f};
        s = wmma4(A0, B0, s);
        s = wmma4(A1, B1, s);
        #pragma unroll
        for (int r = 0; r < 8; ++r)
            if (s[r] > bestv[r]) { bestv[r] = s[r]; besti[r] = code; }
    }

    // butterfly max-reduce within each 16-lane half (token m = r + 8*half)
    #pragma unroll
    for (int r = 0; r < 8; ++r) {
        float bv = bestv[r]; int bi = besti[r];
        #pragma unroll
        for (int m = 8; m >= 1; m >>= 1) {
            float ov = __shfl_xor(bv, m, 32);
            int   oi = __shfl_xor(bi, m, 32);
            if (ov > bv || (ov == bv && oi < bi)) { bv = ov; bi = oi; }
        }
        bestv[r] = bv; besti[r] = bi;
    }
    if (n16 == 0) {                   // lanes 0 and 16 publish 8 tokens each
        #pragma unroll
        for (int r = 0; r < 8; ++r) {
            int tok = r + 8 * half;
            s_idx[wave][tok] = besti[r];
            codes[((size_t)bidx * RVQ_N + stage) * RVQ_T + t0 + tok] = besti[r];
        }
    }
    __syncthreads();

    // ---- gather raw codebook row for this lane's token; losses ----
    int myidx = s_idx[wave][n16];
    const float* cbrow = cbraw + ((size_t)stage * RVQ_CS + myidx) * RVQ_CD;
    float lsum = 0.f;
    if (lane < 16) {
        #pragma unroll
        for (int r = 0; r < RVQ_CD; ++r) { float d = ze[r] - cbrow[r]; lsum += d * d; }
    }
    #pragma unroll
    for (int m = 16; m >= 1; m >>= 1) lsum += __shfl_xor(lsum, m, 32);
    if (lane == 0) {
        const float scl = 1.f / (float)(RVQ_B * RVQ_CD * RVQ_T);
        atomicAdd(&loss[0], lsum * scl);   // commitment
        atomicAdd(&loss[1], lsum * scl);   // codebook (identical in eval mode)
    }

    // ---- out_proj: z_q_out = Wout(512x8) x z_q_i(8x16) + out_b; z_q += ----
    v2f Bo0, Bo1;
    Bo0.x = cbrow[half + 0]; Bo0.y = cbrow[half + 2];
    Bo1.x = cbrow[half + 4]; Bo1.y = cbrow[half + 6];
    const float* ob = out_b + stage * RVQ_D;
    for (int m0 = 0; m0 < RVQ_D; m0 += 16) {
        int dm = m0 + n16;
        v2f Ao0 = *(const v2f*)&s_wout[dm * RVQ_CD + 2 * half];       // (2h,2h+1)
        v2f Ao1 = *(const v2f*)&s_wout[dm * RVQ_CD + 4 + 2 * half];   // (4+2h,5+2h)
        v8f o = {0.f,0.f,0.f,0.f,0.f,0.f,0.f,0.f};
        o = wmma4(Ao0, Bo0, o);
        o = wmma4(Ao1, Bo1, o);
        #pragma unroll
        for (int r = 0; r < 8; ++r) {
            int dd = m0 + r + 8 * half;
            size_t off = (size_t)dd * RVQ_T + t0 + n16;
            float v = o[r] + ob[dd];
            qb[off] = FIRST ? v : (qb[off] + v);
        }
    }
}

// ---------------- host launcher ----------------

extern "C" void kernel_launch(void* const* d_in, const int* in_sizes, int n_in,
                              void* d_out, int out_size, void* d_ws, size_t ws_size,
                              hipStream_t stream) {
    const float* z     = (const float*)d_in[0];
    const float* in_v  = (const float*)d_in[1];
    const float* in_g  = (const float*)d_in[2];
    const float* in_b  = (const float*)d_in[3];
    const float* out_v = (const float*)d_in[4];
    const float* out_g = (const float*)d_in[5];
    const float* out_b = (const float*)d_in[6];
    const float* cb    = (const float*)d_in[7];

    float* out = (float*)d_out;
    const size_t off_codes = (size_t)RVQ_B * RVQ_D * RVQ_T;
    const size_t off_lat   = off_codes + (size_t)RVQ_B * RVQ_N * RVQ_T;
    const size_t off_loss  = off_lat + (size_t)RVQ_B * RVQ_N * RVQ_CD * RVQ_T;
    float* zq    = out;
    int*   codes = (int*)(out + off_codes);
    float* lat   = out + off_lat;
    float* loss  = out + off_loss;

    float* ws     = (float*)d_ws;
    float* wn_in  = ws;                                        // N*16*D (padded)
    float* wn_out = wn_in + (size_t)RVQ_N * 16 * RVQ_D;        // N*D*CD
    float* cbn    = wn_out + (size_t)RVQ_N * RVQ_D * RVQ_CD;   // N*CS*CD (permuted)

    rvq_zero_losses<<<1, 32, 0, stream>>>(loss);
    rvq_prep_in<<<1, RVQ_N * 16, 0, stream>>>(in_v, in_g, wn_in);
    rvq_prep_row8<<<(RVQ_N * RVQ_D + 255) / 256, 256, 0, stream>>>(out_v, out_g, wn_out,
                                                                   RVQ_N * RVQ_D);
    rvq_prep_cbn<<<(RVQ_N * RVQ_CS + 255) / 256, 256, 0, stream>>>(cb, cbn,
                                                                   RVQ_N * RVQ_CS);

    // 9 sequential residual stages; z_q accumulated in place, residual = z - z_q
    rvq_stage<1><<<256, 256, 0, stream>>>(z, zq, codes, lat, loss,
                                          wn_in, wn_out, cbn, cb, in_b, out_b, 0);
    for (int i = 1; i < RVQ_N; ++i) {
        rvq_stage<0><<<256, 256, 0, stream>>>(z, zq, codes, lat, loss,
                                              wn_in, wn_out, cbn, cb, in_b, out_b, i);
    }
}